// Gap_21045339750709
// MI455X (gfx1250) — compile-verified
//
#include <hip/hip_runtime.h>
#include <hip/hip_bf16.h>

// Problem: out(32,64) = sigmoid( mean over (batch, H, W) of x(32,64,224,224) )
// Pure streaming reduction: ~411 MB read @ 23.3 TB/s HBM => ~17.6 us floor.
// Strategy:
//   - grid (C=64, B=32) = 2048 blocks x 256 threads = 16384 wave32 waves.
//   - Each (b,c) chunk is 50176 contiguous f32 = 12544 float4; 256 threads
//     do exactly 49 coalesced non-temporal b128 loads each (streamed once,
//     NT hint keeps the 192MB L2 clean).
//   - Wave reduction via v_wmma_f32_16x16x4_f32 (A = lane partials, B = ones,
//     C = 0) + one shfl_xor(16); block reduction via tiny LDS step.
//   - Partials (64*32 f32) in d_ws, no atomics (deterministic, graph-safe).
//   - Finalize kernel: sum 32 partials/channel, sigmoid, broadcast to 2048 out.

typedef __attribute__((ext_vector_type(2))) float v2f;
typedef __attribute__((ext_vector_type(4))) float f32x4;
typedef __attribute__((ext_vector_type(8))) float v8f;

#define HW      50176     // 224*224
#define HW4     12544     // HW / 4
#define THREADS 256
#define ITERS   49        // HW4 / THREADS (exact)
#define NCHAN   64
#define NBATCH  32

__global__ __launch_bounds__(THREADS) void gap_reduce_kernel(const float* __restrict__ x,
                                                             float* __restrict__ partials) {
    const int c = blockIdx.x;   // channel
    const int b = blockIdx.y;   // batch slice
    const int tid = threadIdx.x;

    // Contiguous 50176-float chunk for this (b, c).
    const f32x4* __restrict__ p =
        (const f32x4*)(x + ((size_t)b * NCHAN + c) * (size_t)HW);

    // Streamed, coalesced, non-temporal b128 loads; 4 independent accumulators.
    float a0 = 0.f, a1 = 0.f, a2 = 0.f, a3 = 0.f;
#pragma unroll 7
    for (int k = 0; k < ITERS; ++k) {
        f32x4 v = __builtin_nontemporal_load(&p[tid + k * THREADS]);
        a0 += v.x; a1 += v.y; a2 += v.z; a3 += v.w;
    }
    float acc = (a0 + a1) + (a2 + a3);

    // ---- Wave32 reduction via WMMA (EXEC is all ones: full block, no branches) ----
    // A (16x4 f32): lanes 0..15 hold (M=lane, K=0) in VGPR0, (M=lane, K=1) in VGPR1;
    // lanes 16..31 hold (M=lane-16, K=2),(K=3). Put partial in slot 0, zero slot 1.
    // B (4x16) = all ones  =>  D[m,n] = p_m + p_{m+16}.
    v2f A; A.x = acc; A.y = 0.f;
    v2f B; B.x = 1.f; B.y = 1.f;
    v8f C = {};
    v8f D = __builtin_amdgcn_wmma_f32_16x16x4_f32(false, A, false, B,
                                                  (short)0, C, false, false);
    // Lane n (0..15): VGPR v = D[v, n]  -> sum_{m=0..7}(p_m + p_{m+16})
    // Lane n (16..31): VGPR v = D[v+8, n-16] -> sum_{m=8..15}(p_m + p_{m+16})
    float w = ((D[0] + D[1]) + (D[2] + D[3])) + ((D[4] + D[5]) + (D[6] + D[7]));
    // Combine the two half-sums: full 32-lane sum in every lane.
    w += __shfl_xor(w, 16, 32);

    // ---- Block reduction (8 waves) ----
    __shared__ float sWave[THREADS / 32];
    const int lane = tid & 31;
    const int wave = tid >> 5;
    if (lane == 0) sWave[wave] = w;
    __syncthreads();
    if (tid == 0) {
        float s = 0.f;
#pragma unroll
        for (int i = 0; i < THREADS / 32; ++i) s += sWave[i];
        partials[c * NBATCH + b] = s;
    }
}

__global__ __launch_bounds__(256) void gap_finalize_kernel(const float* __restrict__ partials,
                                                           float* __restrict__ out) {
    __shared__ float score[NCHAN];
    const int t = threadIdx.x;
    if (t < NCHAN) {
        float s = 0.f;
#pragma unroll
        for (int i = 0; i < NBATCH; ++i) s += partials[t * NBATCH + i];
        const float mean = s * (1.0f / (float)(NBATCH * HW));   // / 1605632
        score[t] = 1.0f / (1.0f + __expf(-mean));
    }
    __syncthreads();
    // Broadcast: out[b, c] = score[c], 32*64 = 2048 elements.
    for (int idx = t; idx < NBATCH * NCHAN; idx += 256)
        out[idx] = score[idx & (NCHAN - 1)];
}

extern "C" void kernel_launch(void* const* d_in, const int* in_sizes, int n_in,
                              void* d_out, int out_size, void* d_ws, size_t ws_size,
                              hipStream_t stream) {
    (void)in_sizes; (void)n_in; (void)out_size; (void)ws_size;
    const float* x = (const float*)d_in[0];
    float* out = (float*)d_out;
    float* partials = (float*)d_ws;   // 64*32 floats = 8 KB, fully rewritten each call

    dim3 grid(NCHAN, NBATCH);
    gap_reduce_kernel<<<grid, THREADS, 0, stream>>>(x, partials);
    gap_finalize_kernel<<<1, 256, 0, stream>>>(partials, out);
}